// HierarchicalResistanceLoss_23905787969836
// MI455X (gfx1250) — compile-verified
//
#include <hip/hip_runtime.h>

typedef float v2f __attribute__((ext_vector_type(2)));
typedef float v4f __attribute__((ext_vector_type(4)));
typedef float v8f __attribute__((ext_vector_type(8)));

#define BLOCKS   2048
#define THREADS  256

// Exact wave32 sum using one V_WMMA_F32_16X16X4_F32.
// A (16x4 f32) layout: lanes 0-15 hold (M=lane, K=0/1) in VGPR0/1,
// lanes 16-31 hold (M=lane-16, K=2/3).  Setting both A VGPRs to the lane
// value v gives A[m][0..1]=v[m], A[m][2..3]=v[m+16].  With B = all-ones:
//   D[m][n] = 2*v[m] + 2*v[m+16]   (independent of n)
// D layout: VGPR r holds row r (lanes 0-15) / row r+8 (lanes 16-31), so the
// per-lane sum of the 8 D VGPRs is 2*sum over half the rows; one xor-16
// shuffle pairs the halves:  total = 0.5 * (s + shfl_xor(s,16)).
__device__ __forceinline__ float wave_sum_wmma(float v) {
    v2f a; a[0] = v;    a[1] = v;
    v2f b; b[0] = 1.0f; b[1] = 1.0f;
    v8f c = {};
    v8f d = __builtin_amdgcn_wmma_f32_16x16x4_f32(
        /*neg_a=*/false, a, /*neg_b=*/false, b,
        /*c_mod=*/(short)0, c, /*reuse_a=*/false, /*reuse_b=*/false);
    float s = ((d[0] + d[1]) + (d[2] + d[3])) + ((d[4] + d[5]) + (d[6] + d[7]));
    s += __shfl_xor(s, 16, 32);
    return s * 0.5f;
}

__global__ void __launch_bounds__(THREADS)
hrl_partial_kernel(const float* __restrict__ pred,
                   const int*   __restrict__ tgt,
                   float*       __restrict__ partial,
                   int n)
{
    const int tid    = blockIdx.x * THREADS + threadIdx.x;
    const int stride = gridDim.x * THREADS;

    float acc = 0.0f;
    for (int i = tid; i < n; i += stride) {
        // Two b128 loads per row; RT temporal hint on purpose: the 144MB
        // working set fits in the 192MB L2, so graph replays hit L2.
        const v4f* p = (const v4f*)(pred + (size_t)i * 8);
        v4f x0 = p[0];
        v4f x1 = p[1];
        int t  = tgt[i];

        float x[8] = { x0[0], x0[1], x0[2], x0[3], x1[0], x1[1], x1[2], x1[3] };

        // max + first-occurrence argmax (matches jnp.argmax semantics)
        float m = x[0]; int am = 0;
        #pragma unroll
        for (int j = 1; j < 8; ++j) {
            if (x[j] > m) { m = x[j]; am = j; }
        }

        // log-sum-exp (max-subtracted) via v_exp_f32 / v_log_f32
        float se = 0.0f;
        #pragma unroll
        for (int j = 0; j < 8; ++j) se += __expf(x[j] - m);

        // gather x[t] with a cndmask chain (no scratch spill)
        float xt = x[0];
        #pragma unroll
        for (int j = 1; j < 8; ++j) xt = (t == j) ? x[j] : xt;

        float ce  = __logf(se) + m - xt;           // -log_softmax[t]
        int   pen = t > am ? (t - am) : (am - t);  // |target - argmax|
        acc += ce * (1.0f + (float)pen);
    }

    // Exact wave reduction via WMMA (EXEC is all-1s here: full block,
    // lanes reconverged after the grid-stride loop).
    float ws = wave_sum_wmma(acc);

    __shared__ float lds[THREADS / 32];
    const int lane = threadIdx.x & 31;
    const int wid  = threadIdx.x >> 5;
    if (lane == 0) lds[wid] = ws;
    __syncthreads();

    if (threadIdx.x == 0) {
        float bsum = 0.0f;
        #pragma unroll
        for (int w = 0; w < THREADS / 32; ++w) bsum += lds[w];
        partial[blockIdx.x] = bsum;
    }
}

__global__ void __launch_bounds__(THREADS)
hrl_final_kernel(const float* __restrict__ partial,
                 float*       __restrict__ out,
                 int nPartials, float invBatch)
{
    float acc = 0.0f;
    // nPartials (2048) is a multiple of THREADS -> uniform trip count,
    // EXEC stays all-1s for the WMMA below.
    for (int i = threadIdx.x; i < nPartials; i += THREADS) acc += partial[i];

    float ws = wave_sum_wmma(acc);

    __shared__ float lds[THREADS / 32];
    const int lane = threadIdx.x & 31;
    const int wid  = threadIdx.x >> 5;
    if (lane == 0) lds[wid] = ws;
    __syncthreads();

    if (threadIdx.x == 0) {
        float s = 0.0f;
        #pragma unroll
        for (int w = 0; w < THREADS / 32; ++w) s += lds[w];
        out[0] = s * invBatch;   // mean
    }
}

extern "C" void kernel_launch(void* const* d_in, const int* in_sizes, int n_in,
                              void* d_out, int out_size, void* d_ws, size_t ws_size,
                              hipStream_t stream)
{
    const float* pred = (const float*)d_in[0];
    const int*   tgt  = (const int*)d_in[1];
    float*       out  = (float*)d_out;
    float*       part = (float*)d_ws;           // BLOCKS floats (8 KB)

    const int n = in_sizes[1];                  // batch = number of targets

    hipLaunchKernelGGL(hrl_partial_kernel, dim3(BLOCKS), dim3(THREADS), 0, stream,
                       pred, tgt, part, n);
    hipLaunchKernelGGL(hrl_final_kernel, dim3(1), dim3(THREADS), 0, stream,
                       part, out, BLOCKS, 1.0f / (float)n);
}